// GraphLayer_1769526526731
// MI455X (gfx1250) — compile-verified
//
#include <hip/hip_runtime.h>

// Problem constants (match reference)
#define KNN_K 16
#define NB    32
#define NP    2048
#define CIN   64
#define COUT  128
#define NTOT  (NB * NP)            // 65536 points
#define TILE  128                  // row/col tile for both matmuls
#define LSTR  68                   // LDS stride for 64-wide feature rows (bank-conflict pad)
#define GSTR  129                  // LDS stride for 128-wide distance tile rows

typedef __attribute__((ext_vector_type(2))) float v2f;
typedef __attribute__((ext_vector_type(8))) float v8f;

// ---------------------------------------------------------------------------
// CDNA5 async global->LDS copy (GLOBAL_LOAD_ASYNC_TO_LDS_B128, GVS mode).
// lds_ptr's low 32 bits are the hardware LDS byte offset (ISA 10.2: generic
// LDS addresses are truncated to addr[31:0]).
// ---------------------------------------------------------------------------
__device__ __forceinline__ void async_copy_b128(void* lds_ptr, const void* g_base,
                                                unsigned byte_off) {
    const unsigned lds_off = (unsigned)(size_t)lds_ptr;
    asm volatile("global_load_async_to_lds_b128 %0, %1, %2 offset:0"
                 :: "v"(lds_off), "v"(byte_off), "s"(g_base) : "memory");
}
__device__ __forceinline__ void async_wait_all() {
    asm volatile("s_wait_asynccnt 0x0" ::: "memory");
}

// ---------------------------------------------------------------------------
// Kernel 0: init pooled = -inf, stats = 0
// ---------------------------------------------------------------------------
__global__ void init_kernel(float* __restrict__ pooled, float* __restrict__ stats) {
    const size_t i = (size_t)blockIdx.x * 256 + threadIdx.x;
    if (i < (size_t)NTOT * CIN) pooled[i] = -__builtin_inff();
    if (i < 512) stats[i] = 0.f;
}

// ---------------------------------------------------------------------------
// Kernel 1: fused KNN. Per block: 128 query rows of one cloud.
// Gram via V_WMMA_F32_16X16X4_F32; tiles staged with async-to-LDS;
// top-16 kept in registers per row.
// ---------------------------------------------------------------------------
__global__ void __launch_bounds__(256) knn_kernel(const float* __restrict__ x,
                                                  int* __restrict__ nbr) {
    extern __shared__ float smem[];
    float* lrow = smem;                     // TILE*LSTR : query rows (resident)
    float* lcol = lrow + TILE * LSTR;       // TILE*LSTR : candidate cols (per iter)
    float* lg   = lcol + TILE * LSTR;       // TILE*GSTR : colnorm - 2*gram
    float* lcn  = lg   + TILE * GSTR;       // TILE      : col squared norms

    const int tid  = threadIdx.x;
    const int wave = tid >> 5;              // 0..7 (wave32)
    const int lane = tid & 31;
    const int half = lane >> 4;             // K-half select for fp32 WMMA frags
    const int l16  = lane & 15;
    const int cloud = blockIdx.x >> 4;      // 16 row-tiles per cloud
    const int row0  = (blockIdx.x & 15) * TILE;
    const float* xc = x + (size_t)cloud * NP * CIN;

    // Stage query row tile (128 x 64) once: async memory->LDS, no VGPR staging.
    for (int i = tid; i < TILE * (CIN / 4); i += 256) {       // exactly 8 per thread
        const int r = i >> 4, c4 = (i & 15) << 2;
        async_copy_b128(&lrow[r * LSTR + c4], xc,
                        (unsigned)(((row0 + r) * CIN + c4) * sizeof(float)));
    }

    float bestd[KNN_K];
    int   besti[KNN_K];
#pragma unroll
    for (int k = 0; k < KNN_K; ++k) { bestd[k] = __builtin_inff(); besti[k] = 0; }

    const int arow = 16 * wave + l16;       // A-matrix row this lane feeds

    for (int ct = 0; ct < NP / TILE; ++ct) {
        const int col0 = ct * TILE;
        __syncthreads();                    // prev scan done before lcol/lg reuse
        for (int i = tid; i < TILE * (CIN / 4); i += 256) {
            const int r = i >> 4, c4 = (i & 15) << 2;
            async_copy_b128(&lcol[r * LSTR + c4], xc,
                            (unsigned)(((col0 + r) * CIN + c4) * sizeof(float)));
        }
        async_wait_all();                   // this wave's async copies (incl. lrow) done
        __syncthreads();                    // -> all waves' copies visible
        if (tid < TILE) {                   // candidate squared norms
            float s = 0.f;
#pragma unroll
            for (int c = 0; c < CIN; ++c) { const float v = lcol[tid * LSTR + c]; s += v * v; }
            lcn[tid] = s;
        }
        __syncthreads();

        // Preload all 16 A fragments (K=64, 4 per WMMA step) for this wave's slab.
        float a0[16], a1[16];
#pragma unroll
        for (int kk = 0; kk < 16; ++kk) {
            const float* ap = &lrow[arow * LSTR + (kk << 2) + (half << 1)];
            a0[kk] = ap[0]; a1[kk] = ap[1];
        }
        for (int ns = 0; ns < 8; ++ns) {    // 8 col sub-tiles of 16
            v8f acc = {0.f, 0.f, 0.f, 0.f, 0.f, 0.f, 0.f, 0.f};
#pragma unroll
            for (int kk = 0; kk < 16; ++kk) {
                const float* bp = &lcol[(ns * 16 + l16) * LSTR + (kk << 2) + (half << 1)];
                v2f a; a.x = a0[kk]; a.y = a1[kk];
                v2f b; b.x = bp[0];  b.y = bp[1];
                acc = __builtin_amdgcn_wmma_f32_16x16x4_f32(false, a, false, b,
                                                            (short)0, acc, false, false);
            }
            const int col = ns * 16 + l16;
            const float cn = lcn[col];
#pragma unroll
            for (int r = 0; r < 8; ++r) {   // C layout: vgpr r -> row r (+8 for hi half)
                const int row = 16 * wave + r + (half << 3);
                lg[row * GSTR + col] = cn - 2.0f * acc[r];
            }
        }
        __syncthreads();

        // Scan: thread t owns row t, register insertion-sorted top-16.
        if (tid < TILE) {
            const int selfc = row0 + tid;
            const float* dr = &lg[tid * GSTR];
            for (int j = 0; j < TILE; ++j) {
                if (col0 + j == selfc) continue;      // loop=False: exclude self
                const float d = dr[j];
                if (d < bestd[KNN_K - 1]) {
                    bestd[KNN_K - 1] = d;
                    besti[KNN_K - 1] = cloud * NP + col0 + j;
#pragma unroll
                    for (int k = KNN_K - 1; k > 0; --k) {
                        if (bestd[k] < bestd[k - 1]) {
                            const float td = bestd[k]; bestd[k] = bestd[k - 1]; bestd[k - 1] = td;
                            const int   ti = besti[k]; besti[k] = besti[k - 1]; besti[k - 1] = ti;
                        }
                    }
                }
            }
        }
    }

    if (tid < TILE) {
        const size_t gp = (size_t)cloud * NP + row0 + tid;
#pragma unroll
        for (int k = 0; k < KNN_K; ++k) nbr[gp * KNN_K + k] = besti[k];
    }
}

// ---------------------------------------------------------------------------
// Kernel 2: scatter-max  pooled[nbr[c][k]] = max(., x[c])  via ordered-int trick
// ---------------------------------------------------------------------------
__device__ __forceinline__ void atomic_max_f32(float* addr, float val) {
    if (val >= 0.f) atomicMax((int*)addr, __float_as_int(val));
    else            atomicMin((unsigned int*)addr, __float_as_uint(val));
}

__global__ void scatter_kernel(const float* __restrict__ x,
                               const int* __restrict__ nbr,
                               float* __restrict__ pooled) {
    const int tid = threadIdx.x;                 // 256 = 4 centers x 64 channels
    const int ch = tid & (CIN - 1);
    const int center = blockIdx.x * 4 + (tid >> 6);
    const float v = x[(size_t)center * CIN + ch];
    const int* ip = nbr + (size_t)center * KNN_K;
#pragma unroll
    for (int k = 0; k < KNN_K; ++k) {
        const int n = ip[k];
        atomic_max_f32(&pooled[(size_t)n * CIN + ch], v);
    }
}

// ---------------------------------------------------------------------------
// Kernel 3: y = pooled @ W^T + b (fp32 WMMA), fused partial BN stats
// ---------------------------------------------------------------------------
__global__ void __launch_bounds__(256) gemm_kernel(const float* __restrict__ pooled,
                                                   const float* __restrict__ W,
                                                   const float* __restrict__ bias,
                                                   float* __restrict__ y,
                                                   float* __restrict__ stats) {
    extern __shared__ float smem[];
    float* lp = smem;                    // TILE*LSTR : activations
    float* lw = lp + TILE * LSTR;        // COUT*LSTR : weights (B operand)
    float* ls = lw + COUT * LSTR;        // 2*COUT    : block-local sum / sumsq

    const int tid = threadIdx.x;
    const int wave = tid >> 5, lane = tid & 31;
    const int half = lane >> 4, l16 = lane & 15;
    const int row0 = blockIdx.x * TILE;

    // W tile: async memory->LDS (no per-element fixup needed).
    for (int i = tid; i < COUT * (CIN / 4); i += 256) {       // exactly 8 per thread
        const int r = i >> 4, c4 = (i & 15) << 2;
        async_copy_b128(&lw[r * LSTR + c4], W,
                        (unsigned)((r * CIN + c4) * sizeof(float)));
    }
    // pooled tile: VGPR path (needs -inf -> 0 fixup, per torch_scatter semantics).
    for (int i = tid; i < TILE * (CIN / 4); i += 256) {
        const int r = i >> 4, c4 = (i & 15) << 2;
        float4 v = *(const float4*)(pooled + (size_t)(row0 + r) * CIN + c4);
        float* d = &lp[r * LSTR + c4];
        d[0] = (__float_as_uint(v.x) == 0xFF800000u) ? 0.f : v.x;
        d[1] = (__float_as_uint(v.y) == 0xFF800000u) ? 0.f : v.y;
        d[2] = (__float_as_uint(v.z) == 0xFF800000u) ? 0.f : v.z;
        d[3] = (__float_as_uint(v.w) == 0xFF800000u) ? 0.f : v.w;
    }
    if (tid < 2 * COUT) ls[tid] = 0.f;
    async_wait_all();
    __syncthreads();

    float a0[16], a1[16];
    const int arow = 16 * wave + l16;
#pragma unroll
    for (int kk = 0; kk < 16; ++kk) {
        const float* ap = &lp[arow * LSTR + (kk << 2) + (half << 1)];
        a0[kk] = ap[0]; a1[kk] = ap[1];
    }
    for (int ns = 0; ns < 8; ++ns) {
        v8f acc = {0.f, 0.f, 0.f, 0.f, 0.f, 0.f, 0.f, 0.f};
#pragma unroll
        for (int kk = 0; kk < 16; ++kk) {
            const float* bp = &lw[(ns * 16 + l16) * LSTR + (kk << 2) + (half << 1)];
            v2f a; a.x = a0[kk]; a.y = a1[kk];
            v2f b; b.x = bp[0];  b.y = bp[1];
            acc = __builtin_amdgcn_wmma_f32_16x16x4_f32(false, a, false, b,
                                                        (short)0, acc, false, false);
        }
        const int col = ns * 16 + l16;
        const float bb = bias[col];
        float s = 0.f, s2 = 0.f;
#pragma unroll
        for (int r = 0; r < 8; ++r) {
            const int row = row0 + 16 * wave + r + (half << 3);
            const float v = acc[r] + bb;
            y[(size_t)row * COUT + col] = v;
            s += v; s2 += v * v;
        }
        atomicAdd(&ls[col], s);          // ds_add_f32
        atomicAdd(&ls[COUT + col], s2);
    }
    __syncthreads();
    if (tid < 2 * COUT) atomicAdd(&stats[tid], ls[tid]);
}

// ---------------------------------------------------------------------------
// Kernel 4: fold BN stats into per-channel scale/shift
// ---------------------------------------------------------------------------
__global__ void finalize_kernel(const float* __restrict__ stats,
                                const float* __restrict__ gamma,
                                const float* __restrict__ beta,
                                float* __restrict__ ss) {
    const int c = threadIdx.x;           // 128
    const float inv_n = 1.0f / (float)NTOT;
    const float mean = stats[c] * inv_n;
    float var = stats[COUT + c] * inv_n - mean * mean;  // biased var
    var = var < 0.f ? 0.f : var;
    const float sc = gamma[c] * rsqrtf(var + 1e-5f);
    ss[c] = sc;
    ss[COUT + c] = beta[c] - mean * sc;
}

// ---------------------------------------------------------------------------
// Kernel 5: normalize + ReLU in place
// ---------------------------------------------------------------------------
__global__ void bnrelu_kernel(float* __restrict__ y, const float* __restrict__ ss) {
    const size_t i = (size_t)blockIdx.x * 256 + threadIdx.x;
    const int c = (int)(i & (COUT - 1));
    const float v = y[i] * ss[c] + ss[COUT + c];
    y[i] = v > 0.f ? v : 0.f;
}

// ---------------------------------------------------------------------------
extern "C" void kernel_launch(void* const* d_in, const int* in_sizes, int n_in,
                              void* d_out, int out_size, void* d_ws, size_t ws_size,
                              hipStream_t stream) {
    (void)in_sizes; (void)n_in; (void)out_size; (void)ws_size;
    const float* x     = (const float*)d_in[0];
    // d_in[1] = batch (int64): sorted, equal clouds of P -> layout is implicit
    const float* W     = (const float*)d_in[2];
    const float* bias  = (const float*)d_in[3];
    const float* gamma = (const float*)d_in[4];
    const float* beta  = (const float*)d_in[5];
    float* y = (float*)d_out;

    char* ws = (char*)d_ws;
    float* stats  = (float*)ws;                                    // 512 floats
    int*   nbr    = (int*)(ws + 2048);                             // NTOT*K ints (4 MB)
    float* pooled = (float*)(ws + 2048 + (size_t)NTOT * KNN_K * sizeof(int)); // NTOT*CIN (16 MB)

    const size_t smem_knn  = (size_t)(TILE * LSTR * 2 + TILE * GSTR + TILE) * sizeof(float); // ~136 KB
    const size_t smem_gemm = (size_t)(TILE * LSTR + COUT * LSTR + 2 * COUT) * sizeof(float); // ~70 KB

    init_kernel<<<((size_t)NTOT * CIN + 255) / 256, 256, 0, stream>>>(pooled, stats);
    knn_kernel<<<NB * (NP / TILE), 256, smem_knn, stream>>>(x, nbr);
    scatter_kernel<<<NTOT / 4, 256, 0, stream>>>(x, nbr, pooled);
    gemm_kernel<<<NTOT / TILE, 256, smem_gemm, stream>>>(pooled, W, bias, y, stats);
    finalize_kernel<<<1, COUT, 0, stream>>>(stats, gamma, beta, stats + 256);
    bnrelu_kernel<<<((size_t)NTOT * COUT) / 256, 256, 0, stream>>>(y, stats + 256);
}